// KabschRMSD_3582002724942
// MI455X (gfx1250) — compile-verified
//
#include <hip/hip_runtime.h>
#include <hip/hip_bf16.h>
#include <math.h>

// ---------------------------------------------------------------------------
// Batched Kabsch RMSD for MI455X (gfx1250, wave32).
//
// Phase 1 (kabsch_gram_kernel): one block (256 thr = 8 waves) per batch sample.
//   Per point i build u_i = (px,py,pz,1, cx,cy,cz,0) and accumulate the 8x8
//   Gram G = sum_i u_i u_i^T with V_WMMA_F32_16X16X4_F32 (K = 4 points/op).
//   A-matrix layout (32-bit, 16x4):  lane l, vgpr j -> M = l&15, K = (l>>4)*2+j
//   B-matrix layout (32-bit, 4x16):  lane l, vgpr j -> N = l&15, K = (l>>4)*2+j
//   => for the symmetric Gram update the SAME register pair is A and B.
//   The per-lane operand value is built branch-free:
//       v = fma(load(lane_ptr), lane_scale, lane_bias)
//   where lane_ptr/scale/bias are loop-invariant per lane (no EXEC juggling
//   in the hot loop; dead lanes load already-resident cachelines and get 0).
//   C/D layout (32-bit 16x16): vgpr r, lane l -> M = r + 8*(l>>4), N = l&15,
//   so the useful 8x8 block lives in lanes 0..7, vgprs 0..7.
//   Waves combine through LDS; lane 0 then runs the closed-form 3x3 Kabsch:
//     A = Spc - Sp Sc^T / N,  Ep = S|p|^2 - |Sp|^2/N,  Ec likewise,
//     sigma = sqrt(eig(A^T A)) (trigonometric closed form),
//     rmsd  = sqrt(max((Ep + Ec - 2*(s1+s2+sign(detA)*s3))/N, 0)).
//
// Phase 2 (mean_kernel): deterministic tree-reduction of the B rmsds -> mean.
//
// Roofline: 2 * 4096 * 2048 * 3 * 4 B ~= 201 MB of reads, ~1.4 FLOP/byte ->
// purely HBM-bound; floor ~= 201 MB / 23.3 TB/s ~= 8.6 us. The WMMA path is
// used because the 17 needed reductions are exactly a rank-K Gram update.
// ---------------------------------------------------------------------------

typedef __attribute__((ext_vector_type(2))) float v2f;
typedef __attribute__((ext_vector_type(8))) float v8f;

#define THREADS 256
#define WAVES   8   // 256 / wave32

__global__ __launch_bounds__(THREADS)
void kabsch_gram_kernel(const float* __restrict__ P,   // [B,N,3]
                        const float* __restrict__ C,   // [B,N,3]
                        float* __restrict__ rmsd_out,  // [B]
                        int N) {
    __shared__ float gram[WAVES][8][8];

    const int b    = blockIdx.x;
    const int lane = threadIdx.x & 31;
    const int wave = threadIdx.x >> 5;

    const float* Pb = P + (size_t)b * (size_t)N * 3u;
    const float* Cb = C + (size_t)b * (size_t)N * 3u;

    const int comp  = lane & 15;         // which component of u this lane holds
    const int khalf = (lane >> 4) << 1;  // K offset from lane half (0 or 2)

    const int ptsPerWave = N / WAVES;    // 256 for N=2048
    const int pt0        = wave * ptsPerWave;

    // ---- loop-invariant per-lane operand descriptor (branch-free hot loop) -
    const float* base = Pb;              // dummy for padding lanes (resident)
    int   compOff = 0;
    float scale   = 0.0f;
    float bias    = 0.0f;
    if (comp < 3)       { base = Pb; compOff = comp;     scale = 1.0f; }
    else if (comp == 3) { bias = 1.0f; }                 // u[3] = 1
    else if (comp < 7)  { base = Cb; compOff = comp - 4; scale = 1.0f; }
    // comp 7..15: scale=0, bias=0 -> contributes 0

    const float* ptr = base + (size_t)(pt0 + khalf) * 3u + (size_t)compOff;

    v8f acc = {};                        // 16x16 f32 accumulator (8 vgprs)

#pragma unroll 4
    for (int it = 0; it < ptsPerWave; it += 4) {
        const float t0 = ptr[0];         // point K = khalf + 0
        const float t1 = ptr[3];         // point K = khalf + 1
        v2f ab;
        ab[0] = fmaf(t0, scale, bias);
        ab[1] = fmaf(t1, scale, bias);
        // D = A x B + C ; A == B == ab  (symmetric rank-4 update)
        acc = __builtin_amdgcn_wmma_f32_16x16x4_f32(
            /*neg_a=*/false, ab, /*neg_b=*/false, ab,
            /*c_mod=*/(short)0, acc, /*reuse_a=*/false, /*reuse_b=*/false);
        ptr += 12;                       // advance 4 points (4 * 3 floats)
    }

    // Park this wave's useful 8x8 block (lanes 0..7 hold N=0..7, vgpr r = M).
    if (lane < 8) {
        float a[8];
#pragma unroll
        for (int r = 0; r < 8; ++r) a[r] = acc[r];
#pragma unroll
        for (int r = 0; r < 8; ++r) gram[wave][r][lane] = a[r];
    }
    __syncthreads();

    // 64 threads: sum across waves for entry (m,n); stash into gram[0].
    if (threadIdx.x < 64) {
        const int m = threadIdx.x >> 3;
        const int n = threadIdx.x & 7;
        float s = 0.0f;
#pragma unroll
        for (int w = 0; w < WAVES; ++w) s += gram[w][m][n];
        gram[0][m][n] = s;
    }
    __syncthreads();

    if (threadIdx.x == 0) {
        const float Nf = (float)N;
        // Pull Gram entries.
        const float spx = gram[0][0][3], spy = gram[0][1][3], spz = gram[0][2][3];
        const float scx = gram[0][3][4], scy = gram[0][3][5], scz = gram[0][3][6];
        const float spp = gram[0][0][0] + gram[0][1][1] + gram[0][2][2];
        const float scc = gram[0][4][4] + gram[0][5][5] + gram[0][6][6];

        const float invN = 1.0f / Nf;
        // Cross-covariance of centered data: A = Spc - Sp Sc^T / N
        float a[3][3];
        const float sp[3] = {spx, spy, spz};
        const float sc[3] = {scx, scy, scz};
#pragma unroll
        for (int j = 0; j < 3; ++j)
#pragma unroll
            for (int k = 0; k < 3; ++k)
                a[j][k] = gram[0][j][4 + k] - sp[j] * sc[k] * invN;

        const float Ep = spp - (spx*spx + spy*spy + spz*spz) * invN;
        const float Ec = scc - (scx*scx + scy*scy + scz*scz) * invN;

        // B = A^T A (symmetric PSD)
        const float b00 = a[0][0]*a[0][0] + a[1][0]*a[1][0] + a[2][0]*a[2][0];
        const float b01 = a[0][0]*a[0][1] + a[1][0]*a[1][1] + a[2][0]*a[2][1];
        const float b02 = a[0][0]*a[0][2] + a[1][0]*a[1][2] + a[2][0]*a[2][2];
        const float b11 = a[0][1]*a[0][1] + a[1][1]*a[1][1] + a[2][1]*a[2][1];
        const float b12 = a[0][1]*a[0][2] + a[1][1]*a[1][2] + a[2][1]*a[2][2];
        const float b22 = a[0][2]*a[0][2] + a[1][2]*a[1][2] + a[2][2]*a[2][2];

        // Closed-form eigenvalues of 3x3 symmetric matrix.
        const float q  = (b00 + b11 + b22) * (1.0f / 3.0f);
        const float p1 = b01*b01 + b02*b02 + b12*b12;
        const float p2 = (b00-q)*(b00-q) + (b11-q)*(b11-q) + (b22-q)*(b22-q)
                       + 2.0f * p1;
        float l1, l2, l3;
        if (p2 <= 1e-30f) {
            l1 = l2 = l3 = q;
        } else {
            const float pp  = sqrtf(p2 * (1.0f / 6.0f));
            const float inv = 1.0f / pp;
            const float c00 = (b00 - q) * inv;
            const float c11 = (b11 - q) * inv;
            const float c22 = (b22 - q) * inv;
            const float c01 = b01 * inv, c02 = b02 * inv, c12 = b12 * inv;
            float r = 0.5f * (c00 * (c11*c22 - c12*c12)
                            - c01 * (c01*c22 - c12*c02)
                            + c02 * (c01*c12 - c11*c02));
            r = fminf(1.0f, fmaxf(-1.0f, r));
            const float phi = acosf(r) * (1.0f / 3.0f);
            l1 = q + 2.0f * pp * cosf(phi);
            l3 = q + 2.0f * pp * cosf(phi + 2.0943951023931953f); // +2*pi/3
            l2 = 3.0f * q - l1 - l3;
        }
        const float s1 = sqrtf(fmaxf(l1, 0.0f));
        const float s2 = sqrtf(fmaxf(l2, 0.0f));
        const float s3 = sqrtf(fmaxf(l3, 0.0f));   // smallest

        const float detA = a[0][0]*(a[1][1]*a[2][2] - a[1][2]*a[2][1])
                         - a[0][1]*(a[1][0]*a[2][2] - a[1][2]*a[2][0])
                         + a[0][2]*(a[1][0]*a[2][1] - a[1][1]*a[2][0]);
        const float d = (detA < 0.0f) ? -1.0f : 1.0f;

        const float trace_term = s1 + s2 + d * s3;
        const float msd = (Ep + Ec - 2.0f * trace_term) * invN;
        rmsd_out[b] = sqrtf(fmaxf(msd, 0.0f));
    }
}

__global__ __launch_bounds__(THREADS)
void mean_kernel(const float* __restrict__ rmsd, float* __restrict__ out, int B) {
    __shared__ float s[THREADS];
    float acc = 0.0f;
    for (int i = threadIdx.x; i < B; i += THREADS) acc += rmsd[i];
    s[threadIdx.x] = acc;
    __syncthreads();
#pragma unroll
    for (int ofs = THREADS / 2; ofs > 0; ofs >>= 1) {
        if (threadIdx.x < (unsigned)ofs) s[threadIdx.x] += s[threadIdx.x + ofs];
        __syncthreads();
    }
    if (threadIdx.x == 0) out[0] = s[0] / (float)B;
}

extern "C" void kernel_launch(void* const* d_in, const int* in_sizes, int n_in,
                              void* d_out, int out_size, void* d_ws, size_t ws_size,
                              hipStream_t stream) {
    const float* coords_pred = (const float*)d_in[0];  // [B,N,3] fp32
    const float* coords      = (const float*)d_in[1];  // [B,N,3] fp32
    float* out = (float*)d_out;                        // scalar fp32

    const int N = 2048;                                // reference fixes N
    const int B = in_sizes[0] / (N * 3);               // 4096

    float* rmsd_ws = (float*)d_ws;                     // B floats of scratch

    kabsch_gram_kernel<<<B, THREADS, 0, stream>>>(coords_pred, coords, rmsd_ws, N);
    mean_kernel<<<1, THREADS, 0, stream>>>(rmsd_ws, out, B);
}